// PointPillarLoss_82205674046060
// MI455X (gfx1250) — compile-verified
//
#include <hip/hip_runtime.h>
#include <math.h>

// PointPillar loss: sparse-gather + focal/smooth-L1 reduction to one scalar.
// Latency-bound; single 1024-thread workgroup, deterministic summation.
// Final cross-thread reduction done with V_WMMA_F32_16X16X4_F32 (ones x B + C).

typedef __attribute__((ext_vector_type(2))) float v2f;
typedef __attribute__((ext_vector_type(8))) float v8f;

__device__ __forceinline__ float focal_term(float p) {
    float om = 1.0f - p;
    return -logf(p) * 0.25f * om * om;   // ALPHA=0.25, GAMMA=2
}

__device__ __forceinline__ float smooth_l1(float x) {
    float ax = fabsf(x);
    return (ax < 1.0f) ? 0.5f * x * x : ax - 0.5f;
}

__global__ __launch_bounds__(1024)
void ppl_loss_kernel(const int*   __restrict__ rt,      // (8,50,2)
                     const int*   __restrict__ ct,      // (8,1000,3)
                     const float* __restrict__ gt,      // (8,50,4)
                     const float* __restrict__ loc,     // (8,2,3,496,432)
                     const float* __restrict__ clf,     // (8,2,3,496,432)
                     const float* __restrict__ anchor,  // (2,)
                     float*       __restrict__ out)
{
    constexpr int B = 8, NB = 50, NBG = 1000, H = 496, W = 432;
    constexpr int HW = H * W;
    constexpr int BSTRIDE = 2 * 3 * HW;            // per-batch stride (A=2, 3 ch)
    constexpr float W_LOC = 2.0f / 400.0f;         // BETA_LOC / (B*n_boxes)
    constexpr float W_CAR = 1.0f / 343.0f;         // 1 / ((B-1)*(n_boxes-1))
    constexpr float W_BG  = 1.0f / 6993.0f;        // 1 / ((B-1)*(n_bg-1))

    const int tid = (int)threadIdx.x;
    const float a0 = anchor[0], a1 = anchor[1];
    const float inv_da = 1.0f / sqrtf(a0 * a0 + a1 * a1);

    float acc = 0.0f;

    // ---- car / localization terms: 400 elements --------------------------
    for (int i = tid; i < B * NB; i += 1024) {
        const int b = i / NB;
        const int x = rt[i * 2 + 0];
        const int y = rt[i * 2 + 1];
        const float g0 = gt[i * 4 + 0];
        const float g1 = gt[i * 4 + 1];
        const float g2 = gt[i * 4 + 2];
        const float g3 = gt[i * 4 + 3];
        const float x_gt = g0 + (g2 - g0) * 0.5f;
        const float y_gt = g1 - (g3 - g1) * 0.5f;
        const int base = b * BSTRIDE + y * W + x;   // [b,0,ch,y,x]
        const float xp = loc[base];                 // loc channel 0
        const float yp = loc[base + HW];            // loc channel 1
        const float cp = clf[base + HW];            // clf channel 1 (car prob)
        const float dx = (x_gt - xp) * inv_da;
        const float dy = (y_gt - yp) * inv_da;
        acc += W_LOC * (smooth_l1(dx) + smooth_l1(dy)) + W_CAR * focal_term(cp);
    }

    // ---- background focal terms: 8000 elements ---------------------------
    for (int i = tid; i < B * NBG; i += 1024) {
        const int b = i / NBG;
        const int x = ct[i * 3 + 1];
        const int y = ct[i * 3 + 2];
        const float p = clf[b * BSTRIDE + y * W + x];  // clf channel 0
        acc += W_BG * focal_term(p);
    }

    __shared__ float part[1024];
    part[tid] = acc;
    __syncthreads();

    // ---- 1024 -> 1 reduction on wave 0 via f32 WMMA ----------------------
    // A = ones(16x4)  =>  D[m][n] = sum_k B[k][n] + C[m][n].
    // Each WMMA consumes 64 partials (bijective lane->(k,n) mapping), so the
    // sum of any row of the accumulator is the total of all values fed in.
    if (tid < 32) {                       // uniform branch: EXEC all-ones in wave 0
        v2f ones; ones[0] = 1.0f; ones[1] = 1.0f;
        v8f c = {};
        #pragma unroll
        for (int t = 0; t < 16; ++t) {
            v2f bv;
            bv[0] = part[t * 64 + tid];
            bv[1] = part[t * 64 + 32 + tid];
            c = __builtin_amdgcn_wmma_f32_16x16x4_f32(
                    /*neg_a=*/false, ones, /*neg_b=*/false, bv,
                    /*c_mod=*/(short)0, c, /*reuse_a=*/false, /*reuse_b=*/false);
        }
        // Row 0 of D lives in c[0] on lanes 0..15 (lanes 16..31 hold row 8,
        // identical values). Reduce within the 16-lane half only.
        float s = c[0];
        s += __shfl_xor(s, 8, 32);
        s += __shfl_xor(s, 4, 32);
        s += __shfl_xor(s, 2, 32);
        s += __shfl_xor(s, 1, 32);
        if (tid == 0) out[0] = s;
    }
}

extern "C" void kernel_launch(void* const* d_in, const int* in_sizes, int n_in,
                              void* d_out, int out_size, void* d_ws, size_t ws_size,
                              hipStream_t stream) {
    (void)in_sizes; (void)n_in; (void)out_size; (void)d_ws; (void)ws_size;
    const int*   rt     = (const int*)d_in[0];   // regression_targets (int32)
    const int*   ct     = (const int*)d_in[1];   // classification_targets (int32)
    const float* gt     = (const float*)d_in[2]; // gt_boxes
    const float* loc    = (const float*)d_in[3]; // loc
    const float* clf    = (const float*)d_in[5]; // clf  (size/occupancy/angle/heading unused)
    const float* anchor = (const float*)d_in[9]; // anchor
    float* out = (float*)d_out;

    ppl_loss_kernel<<<dim3(1), dim3(1024), 0, stream>>>(rt, ct, gt, loc, clf, anchor, out);
}